// EuclideanCodebook_67044439490970
// MI455X (gfx1250) — compile-verified
//
#include <hip/hip_runtime.h>
#include <hip/hip_bf16.h>

#define DIM   512
#define CBOOK 8192
#define NROWS 32768

// GEMM blocking
#define BM 128          // block tile rows (x)
#define BN 256          // block tile cols (embed entries)
#define KC 32           // K chunk staged in LDS
#define PAD 4           // LDS row pad (floats) -> stride 36: bank-conflict-free, 16B aligned
#define SA (KC + PAD)

typedef __attribute__((ext_vector_type(2))) float v2f;
typedef __attribute__((ext_vector_type(8))) float v8f;

// ---------------------------------------------------------------------------
// Reset per-row argmin keys (harness poisons ws once; re-init every call)
// ---------------------------------------------------------------------------
__global__ __launch_bounds__(256) void k_init_keys(unsigned long long* __restrict__ keys) {
    int i = blockIdx.x * 256 + threadIdx.x;
    if (i < NROWS) keys[i] = ~0ull;
}

// ---------------------------------------------------------------------------
// Row squared-norms: one wave32 per row of a [rows, 512] matrix
// ---------------------------------------------------------------------------
__global__ __launch_bounds__(256) void k_row_norms(const float* __restrict__ m,
                                                   float* __restrict__ out) {
    const int wave = threadIdx.x >> 5;
    const int lane = threadIdx.x & 31;
    const int row  = blockIdx.x * 8 + wave;
    const float* p = m + (size_t)row * DIM;
    float s = 0.0f;
#pragma unroll
    for (int i = 0; i < DIM / 32; ++i) {
        float v = p[lane + 32 * i];
        s = fmaf(v, v, s);
    }
#pragma unroll
    for (int off = 16; off; off >>= 1) s += __shfl_xor(s, off, 32);
    if (lane == 0) out[row] = s;
}

// ---------------------------------------------------------------------------
// LDS-staged, register-blocked f32 WMMA GEMM with fused distance + argmin.
//   Block: 256 threads (8 waves) -> 128x256 macro-tile, K staged 32 at a time.
//   Wave : 64x64 register tile = 4x4 grid of 16x16 WMMA accumulators
//          (16 independent v_wmma_f32_16x16x4_f32 per K-chunk -> no C-chain).
// Fragment layout per the CDNA5 ISA (16x4 f32 A / 4x16 f32 B):
//   lanes 0-15 hold K pair {0,1}, lanes 16-31 hold K pair {2,3}; M/N = lane%16.
// ---------------------------------------------------------------------------
__global__ __launch_bounds__(256) void k_dist_wmma(const float* __restrict__ x,
                                                   const float* __restrict__ e,
                                                   const float* __restrict__ x2,
                                                   const float* __restrict__ e2,
                                                   float* __restrict__ dist,
                                                   unsigned long long* __restrict__ keys) {
    __shared__ float As[BM * SA];   // 128 x 36 floats
    __shared__ float Bs[BN * SA];   // 256 x 36 floats

    const int tid  = threadIdx.x;
    const int lane = tid & 31;
    const int wave = tid >> 5;
    const int half = lane >> 4;     // 0: lanes 0-15, 1: lanes 16-31
    const int l16  = lane & 15;
    const int koff = half * 2;      // K sub-offset within a chunk of 4

    const int gm0 = blockIdx.y * BM;        // x rows of this block
    const int gn0 = blockIdx.x * BN;        // codebook entries of this block
    const int m0w = (wave >> 2) * 64;       // wave tile inside block: 2 x 4
    const int n0w = (wave & 3) * 64;

    v8f acc[4][4] = {};

    for (int k0 = 0; k0 < DIM; k0 += KC) {
        __syncthreads();    // previous chunk fully consumed
        // ---- stage A: 128 rows x 32 floats, coalesced b128 ----
#pragma unroll
        for (int it = 0; it < 4; ++it) {
            int q   = tid + 256 * it;           // 1024 float4 slots
            int row = q >> 3, c4 = q & 7;
            float4 v = *(const float4*)&x[(size_t)(gm0 + row) * DIM + k0 + c4 * 4];
            *(float4*)&As[row * SA + c4 * 4] = v;
        }
        // ---- stage B: 256 rows x 32 floats ----
#pragma unroll
        for (int it = 0; it < 8; ++it) {
            int q   = tid + 256 * it;           // 2048 float4 slots
            int row = q >> 3, c4 = q & 7;
            float4 v = *(const float4*)&e[(size_t)(gn0 + row) * DIM + k0 + c4 * 4];
            *(float4*)&Bs[row * SA + c4 * 4] = v;
        }
        __syncthreads();

        // ---- compute: 8 K-chunks of 4, 16 WMMAs each ----
#pragma unroll
        for (int kk = 0; kk < KC; kk += 4) {
            v2f a[4], b[4];
#pragma unroll
            for (int i = 0; i < 4; ++i)
                a[i] = *(const v2f*)&As[(m0w + i * 16 + l16) * SA + kk + koff];
#pragma unroll
            for (int j = 0; j < 4; ++j)
                b[j] = *(const v2f*)&Bs[(n0w + j * 16 + l16) * SA + kk + koff];
#pragma unroll
            for (int i = 0; i < 4; ++i)
#pragma unroll
                for (int j = 0; j < 4; ++j)
                    acc[i][j] = __builtin_amdgcn_wmma_f32_16x16x4_f32(
                        false, a[i], false, b[j], (short)0, acc[i][j], false, false);
        }
    }

    // ---- epilogue: distance, store, fused argmin ----
    float en[4];
#pragma unroll
    for (int j = 0; j < 4; ++j) en[j] = e2[gn0 + n0w + j * 16 + l16];

#pragma unroll
    for (int i = 0; i < 4; ++i) {
#pragma unroll
        for (int r = 0; r < 8; ++r) {
            const int mg = gm0 + m0w + i * 16 + r + half * 8;   // global x row
            const float xm = x2[mg];
            unsigned long long best = ~0ull;
#pragma unroll
            for (int j = 0; j < 4; ++j) {
                const int ng = gn0 + n0w + j * 16 + l16;        // global code index
                float d2 = fmaxf(xm + en[j] - 2.0f * acc[i][j][r], 0.0f);
                dist[(size_t)mg * CBOOK + ng] = -sqrtf(d2);
                // order-preserving key: min key == min distance, ties -> lowest index
                unsigned long long key =
                    ((unsigned long long)__float_as_uint(d2) << 32) | (unsigned)ng;
                if (key < best) best = key;
            }
#pragma unroll
            for (int off = 8; off; off >>= 1) {
                unsigned lo = __shfl_xor((unsigned)best, off, 16);
                unsigned hi = __shfl_xor((unsigned)(best >> 32), off, 16);
                unsigned long long o = ((unsigned long long)hi << 32) | lo;
                if (o < best) best = o;
            }
            if (l16 == 0) atomicMin(&keys[mg], best);
        }
    }
}

// ---------------------------------------------------------------------------
// Gather: quantize[n] = embed[idx[n]]; embed_ind[n] = idx[n]
// ---------------------------------------------------------------------------
__global__ __launch_bounds__(256) void k_gather(const float* __restrict__ e,
                                                const unsigned long long* __restrict__ keys,
                                                float* __restrict__ q,
                                                float* __restrict__ ind) {
    const int row = blockIdx.x;
    const unsigned idx = (unsigned)(keys[row] & 0xFFFFFFFFu);
    const v2f* src = (const v2f*)(e + (size_t)idx * DIM);
    v2f*       dst = (v2f*)(q + (size_t)row * DIM);
    dst[threadIdx.x] = src[threadIdx.x];
    if (threadIdx.x == 0) ind[row] = (float)idx;
}

// ---------------------------------------------------------------------------
extern "C" void kernel_launch(void* const* d_in, const int* in_sizes, int n_in,
                              void* d_out, int out_size, void* d_ws, size_t ws_size,
                              hipStream_t stream) {
    const float* x = (const float*)d_in[0];   // [32768, 512]
    const float* e = (const float*)d_in[1];   // [8192, 512]

    float* out  = (float*)d_out;
    float* q    = out;                              // [N, D]
    float* ind  = out + (size_t)NROWS * DIM;        // [N]
    float* dist = ind + NROWS;                      // [N, C]

    unsigned long long* keys = (unsigned long long*)d_ws;
    float* x2 = (float*)((char*)d_ws + (size_t)NROWS * sizeof(unsigned long long));
    float* e2 = x2 + NROWS;

    hipLaunchKernelGGL(k_init_keys, dim3(NROWS / 256), dim3(256), 0, stream, keys);
    hipLaunchKernelGGL(k_row_norms, dim3(NROWS / 8),   dim3(256), 0, stream, x, x2);
    hipLaunchKernelGGL(k_row_norms, dim3(CBOOK / 8),   dim3(256), 0, stream, e, e2);
    hipLaunchKernelGGL(k_dist_wmma, dim3(CBOOK / BN, NROWS / BM), dim3(256), 0, stream,
                       x, e, x2, e2, dist, keys);
    hipLaunchKernelGGL(k_gather, dim3(NROWS), dim3(256), 0, stream, e, keys, q, ind);
}